// GATModel_66425964200063
// MI455X (gfx1250) — compile-verified
//
#include <hip/hip_runtime.h>
#include <hip/hip_bf16.h>

// ---------------------------------------------------------------------------
// GAT (2-layer) for MI455X / gfx1250.
//   Layer 1: H=8 heads, C=16 -> concat 128.  Layer 2: H=1, C=64.
//   GEMMs via V_WMMA_F32_16X16X4_F32 (exact fp32 on the matrix pipe).
//   Edge phase: 3 memory-bound passes with L2-resident f32 atomics.
// ---------------------------------------------------------------------------

typedef float v2f __attribute__((ext_vector_type(2)));
typedef float v8f __attribute__((ext_vector_type(8)));

#define NEG_SLOPE 0.2f
#define SM_EPS    1e-16f
// ordered-int encoding of -inf for float atomicMax via u32
#define ORD_NEG_INF 0x007FFFFFu

__device__ __forceinline__ unsigned f2ord(float f) {
    unsigned u = __float_as_uint(f);
    return (u & 0x80000000u) ? ~u : (u | 0x80000000u);
}
__device__ __forceinline__ float ord2f(unsigned u) {
    unsigned v = (u & 0x80000000u) ? (u & 0x7FFFFFFFu) : ~u;
    return __uint_as_float(v);
}

// ---------------------------------------------------------------------------
// Y[nrows, NCOLS] = X[nrows, 128] @ W[128, NCOLS]   (K fixed at 128)
// One block = 16 rows; 8 waves, wave w owns columns [16w, 16w+16).
// A tile staged in LDS with b128 loads; NCOLS is compile-time so all row
// strides in the store epilogue are immediate offsets.
// ---------------------------------------------------------------------------
template <int NCOLS>
__global__ void __launch_bounds__(256) wmma_gemm_f32(
    const float* __restrict__ X, const float* __restrict__ W,
    float* __restrict__ Y, int nrows)
{
    __shared__ float As[16][128];
    const int row0 = blockIdx.x * 16;
    const int tid  = threadIdx.x;

    // cooperative b128 load of the 16x128 A tile (512 float4, 2 per thread)
    {
        const float4* X4  = (const float4*)X;
        float4*       As4 = (float4*)&As[0][0];
#pragma unroll
        for (int i = tid; i < 512; i += 256) {
            int r  = i >> 5;           // 32 float4 per row
            int rr = row0 + r;
            float4 v = make_float4(0.f, 0.f, 0.f, 0.f);
            if (rr < nrows) v = X4[(size_t)rr * 32 + (i & 31)];
            As4[i] = v;
        }
    }
    __syncthreads();

    const int wave = tid >> 5;
    const int lane = tid & 31;
    const int col0 = wave * 16;
    if (col0 >= NCOLS) return;                 // whole-wave exit: EXEC stays all-1s

    const int m  = lane & 15;                  // M (A) / N (B,C) index within tile
    const int hi = lane >> 4;                  // half-wave selector

    const float* wp = W + (size_t)col0 + m;    // column base, stride NCOLS (const)
    v8f acc = {};
#pragma unroll 4
    for (int k0 = 0; k0 < 128; k0 += 4) {
        const int ka = k0 + (hi << 1);         // lanes 0-15: K=k0,k0+1; 16-31: K=k0+2,k0+3
        v2f a, b;
        a.x = As[m][ka];
        a.y = As[m][ka + 1];
        b.x = wp[(size_t)ka * NCOLS];
        b.y = wp[(size_t)(ka + 1) * NCOLS];
        acc = __builtin_amdgcn_wmma_f32_16x16x4_f32(
            /*neg_a=*/false, a, /*neg_b=*/false, b,
            /*c_mod=*/(short)0, acc, /*reuse_a=*/false, /*reuse_b=*/false);
    }

    // C layout: VGPR r -> (M = r + 8*hi, N = lane&15)
    const int orow = row0 + (hi ? 8 : 0);
    float* yp = Y + (size_t)orow * NCOLS + col0 + m;
    if (row0 + 16 <= nrows) {                  // hot path: unguarded, imm offsets
#pragma unroll
        for (int r = 0; r < 8; ++r) yp[(size_t)r * NCOLS] = acc[r];
    } else {                                   // tail tile (unused when N%16==0)
#pragma unroll
        for (int r = 0; r < 8; ++r)
            if (orow + r < nrows) yp[(size_t)r * NCOLS] = acc[r];
    }
}

// ---------------------------------------------------------------------------
// a_src[n,h] = sum_c H[n,h,c]*att_src[h,c];  a_dst likewise.
// ---------------------------------------------------------------------------
template <int HEADS, int C>
__global__ void attn_scores(const float* __restrict__ H,
                            const float* __restrict__ att_src,
                            const float* __restrict__ att_dst,
                            float* __restrict__ a_src, float* __restrict__ a_dst,
                            int N)
{
    int t = blockIdx.x * blockDim.x + threadIdx.x;
    if (t >= N * HEADS) return;
    int hd = t & (HEADS - 1);
    const float* hp = H + (size_t)t * C;       // [n][hd][c] contiguous
    const float* as = att_src + hd * C;
    const float* ad = att_dst + hd * C;
    float ss = 0.f, sd = 0.f;
#pragma unroll
    for (int c = 0; c < C; ++c) { float v = hp[c]; ss += v * as[c]; sd += v * ad[c]; }
    a_src[t] = ss; a_dst[t] = sd;
}

__global__ void fill_u32(unsigned* __restrict__ p, unsigned v, int n)
{
    int t = blockIdx.x * blockDim.x + threadIdx.x;
    if (t < n) p[t] = v;
}

// Pass 1: alpha = leakyrelu(a_src[src]+a_dst[dst]); segment max via atomicMax
template <int HEADS>
__global__ void edge_alpha_max(const int* __restrict__ ei, int E, int Etot,
                               const float* __restrict__ a_src,
                               const float* __restrict__ a_dst,
                               float* __restrict__ alpha,
                               unsigned* __restrict__ mmax)
{
    int t = blockIdx.x * blockDim.x + threadIdx.x;
    if (t >= Etot * HEADS) return;
    int e = t / HEADS, hd = t & (HEADS - 1);
    int s, d;
    if (e < E) { s = ei[e]; d = ei[E + e]; } else { s = e - E; d = s; }
    float a = a_src[s * HEADS + hd] + a_dst[d * HEADS + hd];
    a = (a > 0.f) ? a : NEG_SLOPE * a;
    alpha[t] = a;
    atomicMax(&mmax[d * HEADS + hd], f2ord(a));
}

// Pass 2: w = exp(alpha - m[dst]); denom[dst] += w  (w overwrites alpha)
template <int HEADS>
__global__ void edge_exp_denom(const int* __restrict__ ei, int E, int Etot,
                               float* __restrict__ alpha,
                               const unsigned* __restrict__ mmax,
                               float* __restrict__ denom)
{
    int t = blockIdx.x * blockDim.x + threadIdx.x;
    if (t >= Etot * HEADS) return;
    int e = t / HEADS, hd = t & (HEADS - 1);
    int d = (e < E) ? ei[E + e] : (e - E);
    float w = expf(alpha[t] - ord2f(mmax[d * HEADS + hd]));
    alpha[t] = w;
    atomicAdd(&denom[d * HEADS + hd], w);
}

// Pass 3: agg[dst, f] += H[src, f] * w / (denom[dst] + eps)
template <int HEADS, int C>
__global__ void edge_aggregate(const int* __restrict__ ei, int E, int Etot,
                               const float* __restrict__ H,
                               const float* __restrict__ w,
                               const float* __restrict__ denom,
                               float* __restrict__ agg)
{
    constexpr int F = HEADS * C;
    long long t = (long long)blockIdx.x * blockDim.x + threadIdx.x;
    if (t >= (long long)Etot * F) return;
    int e  = (int)(t >> (__builtin_ctz(F)));   // F is a power of two (128 / 64)
    int f  = (int)t & (F - 1);
    int hd = f >> (__builtin_ctz(C));
    int s, d;
    if (e < E) { s = ei[e]; d = ei[E + e]; } else { s = e - E; d = s; }
    float coef = w[e * HEADS + hd] / (denom[d * HEADS + hd] + SM_EPS);
    atomicAdd(&agg[(size_t)d * F + f], H[(size_t)s * F + f] * coef);
}

// out[t] = (in[t] + bias[t & (F-1)]), optional ReLU (in == out allowed)
template <int F, bool RELU>
__global__ void bias_act(const float* __restrict__ in, float* __restrict__ out,
                         const float* __restrict__ bias, int total)
{
    int t = blockIdx.x * blockDim.x + threadIdx.x;
    if (t >= total) return;
    float v = in[t] + bias[t & (F - 1)];
    out[t] = RELU ? fmaxf(v, 0.f) : v;
}

// ---------------------------------------------------------------------------
static inline int cdiv(long long a, int b) { return (int)((a + b - 1) / b); }

extern "C" void kernel_launch(void* const* d_in, const int* in_sizes, int n_in,
                              void* d_out, int out_size, void* d_ws, size_t ws_size,
                              hipStream_t stream)
{
    const float* x        = (const float*)d_in[0];
    const int*   ei       = (const int*)  d_in[1];
    const float* W1       = (const float*)d_in[2];
    const float* att_src1 = (const float*)d_in[3];
    const float* att_dst1 = (const float*)d_in[4];
    const float* b1       = (const float*)d_in[5];
    const float* W2       = (const float*)d_in[6];
    const float* att_src2 = (const float*)d_in[7];
    const float* att_dst2 = (const float*)d_in[8];
    const float* b2       = (const float*)d_in[9];

    constexpr int IN = 128, H1 = 8, C1 = 16, F1 = H1 * C1 /*128*/, C2 = 64;
    const int N    = in_sizes[0] / IN;
    const int E    = in_sizes[1] / 2;
    const int Etot = E + N;   // self-loops appended

    // ---- workspace layout (fp32 words) ----
    size_t off = 0;
    float* base = (float*)d_ws;
    auto take = [&](size_t n) { float* p = base + off; off += n; return p; };
    float*    h1     = take((size_t)N * F1);
    float*    asrc1  = take((size_t)N * H1);
    float*    adst1  = take((size_t)N * H1);
    unsigned* m1     = (unsigned*)take((size_t)N * H1);
    float*    den1   = take((size_t)N * H1);
    float*    w1     = take((size_t)Etot * H1);   // alpha then exp-weights
    float*    agg1   = take((size_t)N * F1);      // also layer-2 input after ReLU
    float*    h2     = take((size_t)N * C2);
    float*    asrc2  = take((size_t)N);
    float*    adst2  = take((size_t)N);
    unsigned* m2     = (unsigned*)take((size_t)N);
    float*    den2   = take((size_t)N);
    float*    w2     = take((size_t)Etot);
    float*    agg2   = take((size_t)N * C2);

    const int B = 256;

    // ================= Layer 1 =================
    wmma_gemm_f32<F1><<<cdiv(N, 16), 256, 0, stream>>>(x, W1, h1, N);
    attn_scores<H1, C1><<<cdiv((long long)N * H1, B), B, 0, stream>>>(
        h1, att_src1, att_dst1, asrc1, adst1, N);

    fill_u32<<<cdiv((long long)N * H1, B), B, 0, stream>>>(m1, ORD_NEG_INF, N * H1);
    fill_u32<<<cdiv((long long)N * H1, B), B, 0, stream>>>((unsigned*)den1, 0u, N * H1);
    fill_u32<<<cdiv((long long)N * F1, B), B, 0, stream>>>((unsigned*)agg1, 0u, N * F1);

    edge_alpha_max<H1><<<cdiv((long long)Etot * H1, B), B, 0, stream>>>(
        ei, E, Etot, asrc1, adst1, w1, m1);
    edge_exp_denom<H1><<<cdiv((long long)Etot * H1, B), B, 0, stream>>>(
        ei, E, Etot, w1, m1, den1);
    edge_aggregate<H1, C1><<<cdiv((long long)Etot * F1, B), B, 0, stream>>>(
        ei, E, Etot, h1, w1, den1, agg1);

    // +b1 then ReLU, in place -> layer-2 input
    bias_act<F1, true><<<cdiv((long long)N * F1, B), B, 0, stream>>>(
        agg1, agg1, b1, N * F1);

    // ================= Layer 2 =================
    wmma_gemm_f32<C2><<<cdiv(N, 16), 256, 0, stream>>>(agg1, W2, h2, N);
    attn_scores<1, C2><<<cdiv((long long)N, B), B, 0, stream>>>(
        h2, att_src2, att_dst2, asrc2, adst2, N);

    fill_u32<<<cdiv((long long)N, B), B, 0, stream>>>(m2, ORD_NEG_INF, N);
    fill_u32<<<cdiv((long long)N, B), B, 0, stream>>>((unsigned*)den2, 0u, N);
    fill_u32<<<cdiv((long long)N * C2, B), B, 0, stream>>>((unsigned*)agg2, 0u, N * C2);

    edge_alpha_max<1><<<cdiv((long long)Etot, B), B, 0, stream>>>(
        ei, E, Etot, asrc2, adst2, w2, m2);
    edge_exp_denom<1><<<cdiv((long long)Etot, B), B, 0, stream>>>(
        ei, E, Etot, w2, m2, den2);
    edge_aggregate<1, C2><<<cdiv((long long)Etot * C2, B), B, 0, stream>>>(
        ei, E, Etot, h2, w2, den2, agg2);

    // final: out = agg2 + b2 (heads=1, mean == identity; no ReLU)
    bias_act<C2, false><<<cdiv((long long)N * C2, B), B, 0, stream>>>(
        agg2, (float*)d_out, b2, N * C2);
}